// SymmetryDistanceLoss_69114613729478
// MI455X (gfx1250) — compile-verified
//
#include <hip/hip_runtime.h>

typedef __attribute__((ext_vector_type(2))) float v2f;
typedef __attribute__((ext_vector_type(8))) float v8f;

#define SGRID 32
#define PTS_PER_BLOCK 128   // 8 waves * 16 points
#define CH_PAD 19           // 18 channels, odd stride to spread LDS banks

__global__ void sym_init_kernel(float* acc) {
    if (threadIdx.x == 0 && blockIdx.x == 0) acc[0] = 0.0f;
}

__global__ __launch_bounds__(256) void sym_loss_kernel(
    const float* __restrict__ outp,     // (B, 6, 4)
    const float* __restrict__ points,   // (B, N, 3)
    const float* __restrict__ closest,  // (B, 32^3, 3)
    float* __restrict__ acc, int N)
{
    __shared__ float T[4][32];                  // 4x18 transform, zero-padded to 4x32
    __shared__ float S[PTS_PER_BLOCK][CH_PAD];  // staged D: 128 points x 18 channels

    const int b = blockIdx.y;
    const int t = threadIdx.x;

    // ---- zero transform matrix (incl. padding columns) ----
    if (t < 128) ((float*)T)[t] = 0.0f;
    __syncthreads();

    // ---- build 4x18 transform matrix: columns = 6 transforms x 3 coords ----
    if (t < 6) {
        const float* o = outp + (size_t)b * 24 + t * 4;
        const float o0 = o[0], o1 = o[1], o2 = o[2], o3 = o[3];
        const int ch0 = t * 3;
        if (t < 3) {
            // plane reflection: refl = (I - 2 n n^T / |n|^2) p - (2 d / |n|^2) n
            const float n[3] = {o0, o1, o2};
            const float inv = 1.0f / (o0 * o0 + o1 * o1 + o2 * o2);
            for (int c = 0; c < 3; ++c) {
                const float s = -2.0f * n[c] * inv;
                for (int k = 0; k < 3; ++k)
                    T[k][ch0 + c] = ((k == c) ? 1.0f : 0.0f) + s * n[k];
                T[3][ch0 + c] = s * o3;         // translation row (K=3, homog 1)
            }
        } else {
            // rotation: rot = R_raw(q) p / ||q||   (matches q p conj(q) / ||q||)
            const float w = o0, x = o1, y = o2, z = o3;
            const float inv = 1.0f / sqrtf(w * w + x * x + y * y + z * z);
            float R[3][3];
            R[0][0] = w*w + x*x - y*y - z*z; R[0][1] = 2.0f*(x*y - w*z); R[0][2] = 2.0f*(x*z + w*y);
            R[1][0] = 2.0f*(x*y + w*z); R[1][1] = w*w - x*x + y*y - z*z; R[1][2] = 2.0f*(y*z - w*x);
            R[2][0] = 2.0f*(x*z - w*y); R[2][1] = 2.0f*(y*z + w*x); R[2][2] = w*w - x*x - y*y + z*z;
            for (int c = 0; c < 3; ++c)
                for (int k = 0; k < 3; ++k)
                    T[k][ch0 + c] = R[c][k] * inv;   // T[3][ch] stays 0
        }
    }
    __syncthreads();

    // ---- WMMA phase: D(16 pts x 16 ch) = A(16x4 homog pts) x B(4x16 transforms) ----
    const int lane = t & 31;
    const int wave = t >> 5;
    const int col  = lane & 15;
    const int hi   = lane >> 4;   // lane half (0: K=0..1, 1: K=2..3)
    const int m    = blockIdx.x * PTS_PER_BLOCK + wave * 16 + col;

    v2f a;
    if (m < N) {
        const float* p = points + ((size_t)b * N + m) * 3;
        if (hi == 0) { a[0] = p[0]; a[1] = p[1]; }   // K=0,1 -> x, y
        else         { a[0] = p[2]; a[1] = 1.0f; }   // K=2,3 -> z, 1 (homogeneous)
    } else {
        a[0] = 0.0f; a[1] = 0.0f;
    }
    // B operand: VGPR g holds K = g + 2*hi (contiguous K halves across lane halves)
    v2f bA, bB;
    bA[0] = T[2 * hi + 0][col];      bA[1] = T[2 * hi + 1][col];
    bB[0] = T[2 * hi + 0][16 + col]; bB[1] = T[2 * hi + 1][16 + col];

    v8f c0 = {};
    // (neg_a, A, neg_b, B, c_mod, C, reuse_a, reuse_b)
    v8f d1 = __builtin_amdgcn_wmma_f32_16x16x4_f32(false, a, false, bA, (short)0, c0, false, false);
    v8f d2 = __builtin_amdgcn_wmma_f32_16x16x4_f32(false, a, false, bB, (short)0, c0, false, false);

    // stage D to LDS: D VGPR g -> point M = g + 8*hi, channel N = col
    const int mloc = wave * 16 + 8 * hi;
#pragma unroll
    for (int g = 0; g < 8; ++g)
        S[mloc + g][col] = d1[g];
    if (col < 2) {
#pragma unroll
        for (int g = 0; g < 8; ++g)
            S[mloc + g][16 + col] = d2[g];
    }
    __syncthreads();

    // ---- gather + distance phase: 768 sym points per block, 3 per thread ----
    float lsum = 0.0f;
    const float* cb = closest + (size_t)b * (SGRID * SGRID * SGRID) * 3;
#pragma unroll
    for (int k = 0; k < 3; ++k) {
        const int s    = t + k * 256;       // 0..767
        const int ml   = s & 127;           // local point
        const int r    = s >> 7;            // transform 0..5
        const int gp   = blockIdx.x * PTS_PER_BLOCK + ml;
        if (gp < N) {
            const float sx = S[ml][r * 3 + 0];
            const float sy = S[ml][r * 3 + 1];
            const float sz = S[ml][r * 3 + 2];
            const int ix = (int)floorf(fminf(fmaxf(sx, 0.0f), 31.0f));
            const int iy = (int)floorf(fminf(fmaxf(sy, 0.0f), 31.0f));
            const int iz = (int)floorf(fminf(fmaxf(sz, 0.0f), 31.0f));
            const int flat = ix * (SGRID * SGRID) + iy * SGRID + iz;
            const float* cp = cb + (size_t)flat * 3;
            const float dx = sx - cp[0], dy = sy - cp[1], dz = sz - cp[2];
            lsum += sqrtf(dx * dx + dy * dy + dz * dz);
        }
    }

    // ---- wave32 reduction, one atomic per wave ----
#pragma unroll
    for (int off = 16; off > 0; off >>= 1)
        lsum += __shfl_xor(lsum, off, 32);
    if (lane == 0)
        atomicAdd(acc, lsum);
}

__global__ void sym_finalize_kernel(const float* acc, float* out, int B, int N) {
    if (threadIdx.x == 0 && blockIdx.x == 0)
        out[0] = acc[0] / ((float)B * (float)N);
}

extern "C" void kernel_launch(void* const* d_in, const int* in_sizes, int n_in,
                              void* d_out, int out_size, void* d_ws, size_t ws_size,
                              hipStream_t stream) {
    const float* outp    = (const float*)d_in[0];   // (B,6,4)
    const float* points  = (const float*)d_in[1];   // (B,N,3)
    const float* closest = (const float*)d_in[2];   // (B,32768,3)

    const int B = in_sizes[0] / 24;
    const int N = in_sizes[1] / (3 * B);

    float* acc = (float*)d_ws;

    sym_init_kernel<<<1, 1, 0, stream>>>(acc);
    dim3 grid((N + PTS_PER_BLOCK - 1) / PTS_PER_BLOCK, B);
    sym_loss_kernel<<<grid, 256, 0, stream>>>(outp, points, closest, acc, N);
    sym_finalize_kernel<<<1, 1, 0, stream>>>(acc, (float*)d_out, B, N);
}